// Cache_65627100283720
// MI455X (gfx1250) — compile-verified
//
#include <hip/hip_runtime.h>
#include <hip/hip_bf16.h>

// ---------------------------------------------------------------------------
// Shapes (from reference): QL=4, L=32, B=16, H=512, N=128, DK=512, LH=16384
//   query     : [4,32,16,512]  fp32   (elem (i,l,b,h) at ((i*32+l)*16+b)*512+h)
//   keys      : [128,16,512]   fp32
//   values    : unused (attended output is discarded by the reference)
//   W_summary : [512,16384]    fp32 row-major
//   b_summary : [512]          fp32
// Outputs: attention [64,1,128] fp32 (8192) ++ topk_indices [16,64] (1024, as f32)
//
// Roofline: the summary GEMM streams 33.5 MB of fp32 W (+4.2 MB query, L2-
// resident) -> ~1.6 us at 23.3 TB/s. fp32 WMMA keeps full precision and the
// kernel stays HBM-bound; down-converting would save zero HBM bytes.
// ---------------------------------------------------------------------------

typedef __attribute__((ext_vector_type(2))) float v2f;
typedef __attribute__((ext_vector_type(8))) float v8f;

// ---------------------------------------------------------------------------
// Kernel 1: summary GEMM partials with V_WMMA_F32_16X16X4_F32.
// grid=(32 N-tiles, 8 K-splits), block=128 (4 waves). Wave w computes the
// 16x16 f32 tile [rows 16w..16w+15] x [cols 16*bx..16*bx+15] over K-slice
// [2048*by, 2048*by+2048).
//
// The K-slice is 4 contiguous l-chunks of 512 floats in both operands, so we
// hoist chunk math and stream two flat pointers; all fragment loads become
// immediate-offset global_load_b64s (A: lane holds row m=lane&15, k-pair
// {2*hi, 2*hi+1}; B: lane holds col n=lane&15, same k-pair).
// P layout: P[(by*32+bx)*1024 + m*16 + nl]
// ---------------------------------------------------------------------------
__global__ __launch_bounds__(128) void summary_gemm_partial(
    const float* __restrict__ q, const float* __restrict__ W,
    float* __restrict__ P) {
  const int lane = threadIdx.x & 31;
  const int wave = threadIdx.x >> 5;
  const int lo   = lane & 15;
  const int hi   = lane >> 4;

  // A-fragment row (same for both lane halves): m = wave*16 + lo
  const int m = wave * 16 + lo;                    // global output row 0..63
  const int qi = m >> 4, qb = m & 15;              // r = i*16 + b
  const int rowbase = qi * 262144 + qb * 512;

  // B-fragment column: d = 16*bx + lo
  const int d = (int)blockIdx.x * 16 + lo;
  const int k0 = (int)blockIdx.y * 2048;
  const int l0 = (int)blockIdx.y * 4;              // first l-chunk of slice

  v8f acc = {0.f, 0.f, 0.f, 0.f, 0.f, 0.f, 0.f, 0.f};

  for (int c = 0; c < 4; ++c) {                    // l-chunks of 512
    const float* pA = q + rowbase + (l0 + c) * 8192 + 2 * hi;
    const float* pB = W + d * 16384 + k0 + c * 512 + 2 * hi;
    for (int kk = 0; kk < 512; kk += 64) {         // 16 WMMAs / iteration
      __builtin_prefetch((const void*)(pB + 256), 0, 3);
#pragma unroll
      for (int t = 0; t < 16; ++t) {
        v2f a = *(const v2f*)(pA + 4 * t);         // imm offsets 0..240B
        v2f b = *(const v2f*)(pB + 4 * t);
        acc = __builtin_amdgcn_wmma_f32_16x16x4_f32(
            false, a, false, b, (short)0, acc, false, false);
      }
      pA += 64;
      pB += 64;
    }
  }

  // C/D layout: VGPR v -> M = v + 8*hi ; N = lo
  float* out = P + ((int)blockIdx.y * 32 + (int)blockIdx.x) * 1024;
#pragma unroll
  for (int v = 0; v < 8; ++v) {
    const int mm = wave * 16 + v + 8 * hi;
    out[mm * 16 + lo] = acc[v];
  }
}

// ---------------------------------------------------------------------------
// Kernel 2: reduce the 8 K-split partials + bias -> q2[64][512]
// ---------------------------------------------------------------------------
__global__ __launch_bounds__(256) void reduce_bias(
    const float* __restrict__ P, const float* __restrict__ bias,
    float* __restrict__ q2) {
  const int tid = (int)blockIdx.x * 256 + (int)threadIdx.x;
  if (tid >= 64 * 512) return;
  const int r = tid >> 9, d = tid & 511;
  const int nx = d >> 4, nl = d & 15;
  float s = bias[d];
#pragma unroll
  for (int ky = 0; ky < 8; ++ky)
    s += P[((ky * 32 + nx) * 64 + r) * 16 + nl];
  q2[tid] = s;
}

// ---------------------------------------------------------------------------
// Kernel 3: scores[r][n] = dot(q2[r], keys[n, b=r%16]) / sqrt(512)
// Memory-bound on the 16 MB keys read; float4 streaming loads.
// ---------------------------------------------------------------------------
__global__ __launch_bounds__(128) void scores_kernel(
    const float* __restrict__ q2, const float* __restrict__ keys,
    float* __restrict__ sc) {
  const int tid = (int)blockIdx.x * 128 + (int)threadIdx.x;
  if (tid >= 64 * 128) return;
  const int r = tid >> 7, n = tid & 127;
  const int b = r & 15;
  const float4* qa = (const float4*)(q2 + r * 512);
  const float4* ka = (const float4*)(keys + (n * 16 + b) * 512);
  float s = 0.f;
#pragma unroll 8
  for (int dd = 0; dd < 128; ++dd) {
    const float4 x = qa[dd], y = ka[dd];
    s += x.x * y.x + x.y * y.y + x.z * y.z + x.w * y.w;
  }
  sc[tid] = s * 0.04419417382415922f;              // 1/sqrt(512)
}

// ---------------------------------------------------------------------------
// Kernel 4: per-row softmax (-> attention) and top-16 indices.
// grid=64 rows, block=128 (one thread per cached segment n).
// Top-k uses strict '>' so lowest index wins ties (matches jax.lax.top_k).
// ---------------------------------------------------------------------------
__global__ __launch_bounds__(128) void softmax_topk(
    const float* __restrict__ sc, float* __restrict__ out) {
  __shared__ float buf[128];
  __shared__ float wsm[128];
  const int r = (int)blockIdx.x;
  const int t = (int)threadIdx.x;

  const float s = sc[r * 128 + t];
  buf[t] = s;
  __syncthreads();
  for (int off = 64; off > 0; off >>= 1) {
    if (t < off) buf[t] = fmaxf(buf[t], buf[t + off]);
    __syncthreads();
  }
  const float mx = buf[0];
  __syncthreads();

  const float e = expf(s - mx);
  buf[t] = e;
  __syncthreads();
  for (int off = 64; off > 0; off >>= 1) {
    if (t < off) buf[t] += buf[t + off];
    __syncthreads();
  }
  const float wgt = e / buf[0];
  out[r * 128 + t] = wgt;                          // attention [64,1,128]
  wsm[t] = wgt;
  __syncthreads();

  if (t == 0) {
    for (int sel = 0; sel < 16; ++sel) {
      float best = -1.0f;
      int bi = 0;
      for (int n = 0; n < 128; ++n)
        if (wsm[n] > best) { best = wsm[n]; bi = n; }
      wsm[bi] = -2.0f;
      out[8192 + sel * 64 + r] = (float)bi;        // topk_indices.T [16,64]
    }
  }
}

// ---------------------------------------------------------------------------
extern "C" void kernel_launch(void* const* d_in, const int* in_sizes, int n_in,
                              void* d_out, int out_size, void* d_ws,
                              size_t ws_size, hipStream_t stream) {
  const float* query = (const float*)d_in[0];
  const float* keys  = (const float*)d_in[1];
  // d_in[2] = values: only feeds a discarded einsum in the reference -> skip.
  const float* Wsum  = (const float*)d_in[3];
  const float* bsum  = (const float*)d_in[4];

  float* ws = (float*)d_ws;
  float* P  = ws;                     // 32*8*64*16 = 262144 floats (1 MB)
  float* q2 = ws + 262144;            // 64*512    =  32768 floats
  float* sc = ws + 262144 + 32768;    // 64*128    =   8192 floats

  dim3 g1(32, 8), b1(128);
  summary_gemm_partial<<<g1, b1, 0, stream>>>(query, Wsum, P);
  reduce_bias<<<(64 * 512 + 255) / 256, 256, 0, stream>>>(P, bsum, q2);
  scores_kernel<<<(64 * 128 + 127) / 128, 128, 0, stream>>>(q2, keys, sc);
  softmax_topk<<<64, 128, 0, stream>>>(sc, (float*)d_out);
}